// HI4B1C_codebook_69587060130221
// MI455X (gfx1250) — compile-verified
//
#include <hip/hip_runtime.h>
#include <stdint.h>

// HI4B1C nearest-grid quantize + nibble-pack + dequant for MI455X (gfx1250).
//
// Memory-bound kernel: 64MB in, 200MB out  =>  ~11.4us floor at 23.3 TB/s.
// Data path: global_load_async_to_lds_b128 (CDNA5 async mover, ASYNCcnt) ->
// LDS -> ds_load_b128 -> VALU closed-form argmax -> non-temporal b128 stores
// (output is write-once and larger than the 192MB L2, so th:NT).

typedef float v4f __attribute__((ext_vector_type(4)));
typedef int   v4i __attribute__((ext_vector_type(4)));

#define TPB 256
#define EPT 8                       // elements per thread = one packed word
#define ELEMS_PER_BLOCK (TPB * EPT) // 2048 floats = 8KB tile

__global__ __launch_bounds__(TPB) void hi4b1c_quant_pack_kernel(
    const float* __restrict__ X,
    float* __restrict__ out,
    int nElems,   // R*C
    int nWords)   // nElems / 8
{
    __shared__ float smem[ELEMS_PER_BLOCK];

    const int tid  = threadIdx.x;
    const int word = blockIdx.x * TPB + tid;          // packed-word index
    const int blockElemBase = blockIdx.x * ELEMS_PER_BLOCK;

    float x[EPT];

    if (blockElemBase + ELEMS_PER_BLOCK <= nElems) {
        // ---- fast path: stage the 8KB block tile via the async data mover ----
        // Wave w stages bytes [w*1024, w*1024+1024) of the tile; each lane
        // issues two b128 async ops (16B each, lanes contiguous -> 512B/op).
        const int wave = tid >> 5;
        const int lane = tid & 31;
        const int rel0 = wave * 1024 + lane * 16;     // byte offset in tile

        uint64_t g0 = (uint64_t)((const char*)X + (size_t)blockElemBase * 4 + rel0);
        uint64_t g1 = g0 + 512;
        // low 32 bits of the flat address of an LDS object == LDS byte offset
        uint32_t l0 = (uint32_t)(uintptr_t)(&smem[0]) + (uint32_t)rel0;
        uint32_t l1 = l0 + 512;

        asm volatile("global_load_async_to_lds_b128 %0, %1, off"
                     :: "v"(l0), "v"(g0) : "memory");
        asm volatile("global_load_async_to_lds_b128 %0, %1, off"
                     :: "v"(l1), "v"(g1) : "memory");
        // ASYNCcnt==0 => data landed in LDS; wave consumes only its own
        // staged region, so no workgroup barrier is required.
        asm volatile("s_wait_asynccnt 0" ::: "memory");

        const v4f* s4 = reinterpret_cast<const v4f*>(smem) + tid * 2;
        v4f a = s4[0];   // ds_load_b128
        v4f b = s4[1];
        x[0] = a[0]; x[1] = a[1]; x[2] = a[2]; x[3] = a[3];
        x[4] = b[0]; x[5] = b[1]; x[6] = b[2]; x[7] = b[3];
    } else {
        // ---- tail path (not hit for 4096x4096, kept for generality) ----
        #pragma unroll
        for (int p = 0; p < EPT; ++p) {
            int e = word * EPT + p;
            x[p] = (e < nElems) ? X[e] : 0.0f;
        }
    }

    if (word >= nWords) return;

    // Nearest grid point among {k - 7.5 : k=0..15}; argmax ties (x at exact
    // integer midpoints) resolve to the lower k, which ceil(x+7) reproduces.
    int k[EPT];
    float v[EPT];
    #pragma unroll
    for (int p = 0; p < EPT; ++p) {
        int t = (int)__builtin_ceilf(x[p] + 7.0f);
        t = t < 0 ? 0 : (t > 15 ? 15 : t);            // v_med3-able clamp
        k[p] = t;
        v[p] = (float)t - 7.5f;                        // == grid[t], exact
    }

    // torch interleaved nibble order: position p -> shift {28,12,24,8,20,4,16,0}
    uint32_t packed =
        ((uint32_t)k[0] << 28) | ((uint32_t)k[1] << 12) |
        ((uint32_t)k[2] << 24) | ((uint32_t)k[3] <<  8) |
        ((uint32_t)k[4] << 20) | ((uint32_t)k[5] <<  4) |
        ((uint32_t)k[6] << 16) | ((uint32_t)k[7]);

    // Output layout (flat concat, in return order):
    //   vals   [nElems] f32   @ 0
    //   idx    [nElems] i32   @ nElems
    //   packed [nWords] i32   @ 2*nElems
    //   dequant[nElems] f32   @ 2*nElems + nWords   (== vals bit-exactly)
    float* valsOut = out;
    int*   idxOut  = (int*)out + (size_t)nElems;
    int*   pkOut   = (int*)out + (size_t)2 * nElems;
    float* deqOut  = out + (size_t)2 * nElems + (size_t)nWords;

    v4f v0 = { v[0], v[1], v[2], v[3] };
    v4f v1 = { v[4], v[5], v[6], v[7] };
    v4i i0 = { k[0], k[1], k[2], k[3] };
    v4i i1 = { k[4], k[5], k[6], k[7] };

    const size_t e = (size_t)word * EPT;
    // Write-once streaming output > L2 capacity: non-temporal b128 stores.
    __builtin_nontemporal_store(v0, reinterpret_cast<v4f*>(valsOut + e));
    __builtin_nontemporal_store(v1, reinterpret_cast<v4f*>(valsOut + e + 4));
    __builtin_nontemporal_store(i0, reinterpret_cast<v4i*>(idxOut + e));
    __builtin_nontemporal_store(i1, reinterpret_cast<v4i*>(idxOut + e + 4));
    __builtin_nontemporal_store((int)packed, pkOut + word);
    __builtin_nontemporal_store(v0, reinterpret_cast<v4f*>(deqOut + e));
    __builtin_nontemporal_store(v1, reinterpret_cast<v4f*>(deqOut + e + 4));
}

extern "C" void kernel_launch(void* const* d_in, const int* in_sizes, int n_in,
                              void* d_out, int out_size, void* d_ws, size_t ws_size,
                              hipStream_t stream) {
    (void)n_in; (void)d_ws; (void)ws_size; (void)out_size;
    const float* X = (const float*)d_in[0];   // [4096,4096,1] f32
    // d_in[1] (grid) is the fixed half-integer grid; folded into closed form.
    int nElems = in_sizes[0];                 // R*C (d==1)
    int nWords = nElems / 8;
    int blocks = (nWords + TPB - 1) / TPB;    // 8192 blocks for 4096x4096
    hi4b1c_quant_pack_kernel<<<blocks, TPB, 0, stream>>>(
        X, (float*)d_out, nElems, nWords);
}